// Block_71554155151901
// MI455X (gfx1250) — compile-verified
//
#include <hip/hip_runtime.h>
#include <hip/hip_bf16.h>

typedef float v2f __attribute__((ext_vector_type(2)));
typedef float v8f __attribute__((ext_vector_type(8)));

#define WMMA_F32(acc, a, b) \
  __builtin_amdgcn_wmma_f32_16x16x4_f32(false, (a), false, (b), (short)0, (acc), false, false)

__device__ __forceinline__ float silu_f(float x) { return x / (1.f + __expf(-x)); }

// float atomic-max via ordered-int trick; sentinel init = 0xFFFFFFFF bytes (negative NaN,
// int -1, unsigned max) so the first real value always wins.
__device__ __forceinline__ void atomic_max_f32(float* addr, float val) {
  if (val >= 0.f) atomicMax((int*)addr, __float_as_int(val));
  else            atomicMin((unsigned int*)addr, (unsigned int)__float_as_int(val));
}

// One wave computes a 16x16 fp32 tile D += A(16xK from LDS, row-major lda) * B(KxNB global,
// cols [col0,col0+16)) using V_WMMA_F32_16X16X4_F32 over K/4 steps.
// 32-bit A layout: lanes 0-15 -> M=0..15 K={k0,k0+1}; lanes 16-31 -> M=0..15 K={k0+2,k0+3}.
// B mirrored with N=lane&15.
__device__ __forceinline__ v8f wmma_gemm_tile(const float* ldsA, int lda,
                                              const float* __restrict__ W, int nb,
                                              int col0, int K, v8f acc) {
  const int lane = threadIdx.x & 31;
  const int half = lane >> 4;
  const int mr   = lane & 15;
  for (int k0 = 0; k0 < K; k0 += 4) {
    const int ka = k0 + 2 * half;
    v2f a, b;
    a.x = ldsA[mr * lda + ka];
    a.y = ldsA[mr * lda + ka + 1];
    b.x = W[ka * nb + col0 + mr];
    b.y = W[(ka + 1) * nb + col0 + mr];
    acc = WMMA_F32(acc, a, b);
  }
  return acc;
}

// LayerNorm(+gamma,beta) then SiLU over a 16x64 LDS tile, in place.
__device__ __forceinline__ void ln_silu_16x64(float* buf, const float* __restrict__ g,
                                              const float* __restrict__ be,
                                              float* mu, float* rs, int tid, int nthr) {
  if (tid < 16) {
    float m = 0.f;
    for (int j = 0; j < 64; ++j) m += buf[tid * 64 + j];
    m *= (1.f / 64.f);
    float v = 0.f;
    for (int j = 0; j < 64; ++j) { float d = buf[tid * 64 + j] - m; v += d * d; }
    mu[tid] = m;
    rs[tid] = rsqrtf(v * (1.f / 64.f) + 1e-6f);
  }
  __syncthreads();
  for (int i = tid; i < 16 * 64; i += nthr) {
    int e = i >> 6, j = i & 63;
    float x = (buf[i] - mu[e]) * rs[e] * g[j] + be[j];
    buf[i] = silu_f(x);
  }
  __syncthreads();
}

// ---------------- Kernel 1: per-edge radial TP weights, scal, alpha MLP, seg-max ----------
__global__ __launch_bounds__(128) void edge_prep_kernel(
    const float* __restrict__ xlen, const float* __restrict__ edge_sh,
    const int* __restrict__ edge_dst,
    const float* __restrict__ tp2_w1, const float* __restrict__ tp2_b1,
    const float* __restrict__ tp2_w2, const float* __restrict__ tp2_b2,
    const float* __restrict__ a_w1, const float* __restrict__ a_b1,
    const float* __restrict__ a_g1, const float* __restrict__ a_be1,
    const float* __restrict__ a_w2, const float* __restrict__ a_b2,
    const float* __restrict__ a_g2, const float* __restrict__ a_be2,
    const float* __restrict__ a_w3, const float* __restrict__ a_b3,
    float* __restrict__ ws_alpha, float* __restrict__ ws_scal,
    float* __restrict__ node_max, long long E) {
  __shared__ float xe[16 * 64];
  __shared__ float buf1[16 * 64];
  __shared__ float buf2[16 * 64];
  __shared__ float mu[16], rs[16];

  const int tid  = threadIdx.x;
  const int wid  = tid >> 5;
  const int lane = tid & 31;
  const int half = lane >> 4;
  const int mr   = lane & 15;
  const long long e0 = (long long)blockIdx.x * 16;

  for (int i = tid; i < 16 * 64; i += 128) {
    int e = i >> 6, c = i & 63;
    long long ee = e0 + e; if (ee >= E) ee = E - 1;
    xe[i] = xlen[ee * 64 + c];
  }
  __syncthreads();

  // radial = silu(xe @ tp2_w1 + b1) @ tp2_w2 + b2   -> buf2 [16][H*S]
  {
    const int col0 = wid * 16;
    v8f acc = {0.f,0.f,0.f,0.f,0.f,0.f,0.f,0.f};
    acc = wmma_gemm_tile(xe, 64, tp2_w1, 64, col0, 64, acc);
#pragma unroll
    for (int r = 0; r < 8; ++r) {
      int m = r + 8 * half, n = col0 + mr;
      buf1[m * 64 + n] = silu_f(acc[r] + tp2_b1[n]);
    }
  }
  __syncthreads();
  {
    const int col0 = wid * 16;
    v8f acc = {0.f,0.f,0.f,0.f,0.f,0.f,0.f,0.f};
    acc = wmma_gemm_tile(buf1, 64, tp2_w2, 64, col0, 64, acc);
#pragma unroll
    for (int r = 0; r < 8; ++r) {
      int m = r + 8 * half, n = col0 + mr;
      buf2[m * 64 + n] = acc[r] + tp2_b2[n];
    }
  }
  __syncthreads();
  // scal[e][h] = sum_s radial[e][h*16+s] * edge_sh[e][s]
  if (tid < 64) {
    int e = tid >> 2, h = tid & 3;
    long long ee = e0 + e;
    if (ee < E) {
      float s = 0.f;
#pragma unroll
      for (int si = 0; si < 16; ++si)
        s += buf2[e * 64 + h * 16 + si] * edge_sh[ee * 16 + si];
      ws_scal[ee * 4 + h] = s;
    }
  }
  // alpha = mlp3(xe): GEMM -> LN+SiLU -> GEMM -> LN+SiLU -> GEMM(64x4)
  {
    const int col0 = wid * 16;
    v8f acc = {0.f,0.f,0.f,0.f,0.f,0.f,0.f,0.f};
    acc = wmma_gemm_tile(xe, 64, a_w1, 64, col0, 64, acc);
#pragma unroll
    for (int r = 0; r < 8; ++r) {
      int m = r + 8 * half, n = col0 + mr;
      buf1[m * 64 + n] = acc[r] + a_b1[n];
    }
  }
  __syncthreads();
  ln_silu_16x64(buf1, a_g1, a_be1, mu, rs, tid, 128);
  {
    const int col0 = wid * 16;
    v8f acc = {0.f,0.f,0.f,0.f,0.f,0.f,0.f,0.f};
    acc = wmma_gemm_tile(buf1, 64, a_w2, 64, col0, 64, acc);
#pragma unroll
    for (int r = 0; r < 8; ++r) {
      int m = r + 8 * half, n = col0 + mr;
      buf2[m * 64 + n] = acc[r] + a_b2[n];
    }
  }
  __syncthreads();
  ln_silu_16x64(buf2, a_g2, a_be2, mu, rs, tid, 128);
  if (wid == 0) {  // 64x4 tail GEMM: zero-pad B cols 4..15
    v8f acc = {0.f,0.f,0.f,0.f,0.f,0.f,0.f,0.f};
    for (int k0 = 0; k0 < 64; k0 += 4) {
      int ka = k0 + 2 * half;
      v2f a, b;
      a.x = buf2[mr * 64 + ka];
      a.y = buf2[mr * 64 + ka + 1];
      b.x = (mr < 4) ? a_w3[ka * 4 + mr] : 0.f;
      b.y = (mr < 4) ? a_w3[(ka + 1) * 4 + mr] : 0.f;
      acc = WMMA_F32(acc, a, b);
    }
    if (mr < 4) {
#pragma unroll
      for (int r = 0; r < 8; ++r) {
        int m = r + 8 * half;
        long long ee = e0 + m;
        if (ee < E) {
          float av = acc[r] + a_b3[mr];
          ws_alpha[ee * 4 + mr] = av;
          atomic_max_f32(&node_max[(long long)edge_dst[ee] * 4 + mr], av);
        }
      }
    }
  }
}

// ---------------- Kernel 2: softmax denominators -------------------------------------------
__global__ __launch_bounds__(256) void softmax_sum_kernel(
    const float* __restrict__ ws_alpha, const int* __restrict__ edge_dst,
    const float* __restrict__ node_max, float* __restrict__ node_sum, long long EH) {
  long long i = (long long)blockIdx.x * blockDim.x + threadIdx.x;
  if (i >= EH) return;
  long long e = i >> 2; int h = (int)(i & 3);
  long long ni = (long long)edge_dst[e] * 4 + h;
  atomicAdd(&node_sum[ni], __expf(ws_alpha[i] - node_max[ni]));
}

// ---------------- Kernel 3: pre_lin GEMM, edge_scalar chain, node scatter ------------------
__global__ __launch_bounds__(256) void edge_main_kernel(
    const float* __restrict__ node_in, const float* __restrict__ xlen,
    const int* __restrict__ edge_src, const int* __restrict__ edge_dst,
    const float* __restrict__ pre_lin_w, const float* __restrict__ lin_edge_w,
    const float* __restrict__ s_w1, const float* __restrict__ s_b1,
    const float* __restrict__ s_g1, const float* __restrict__ s_be1,
    const float* __restrict__ s_w2, const float* __restrict__ s_b2,
    const float* __restrict__ s_g2, const float* __restrict__ s_be2,
    const float* __restrict__ s_w3, const float* __restrict__ s_b3,
    const float* __restrict__ ws_alpha, const float* __restrict__ ws_scal,
    const float* __restrict__ node_max, const float* __restrict__ node_sum,
    float* __restrict__ node_fea, float* __restrict__ out_edge, long long E) {
  __shared__ float Acat[16 * 320];   // gathered [src | dst | len] rows
  __shared__ float msg[16 * 128];
  __shared__ float buf1[16 * 64];
  __shared__ float buf2[16 * 64];
  __shared__ float coefA[16 * 4];    // scal*alpha  (edge_out coefficient)
  __shared__ float coefN[16 * 4];    // scal*asm    (node-accum coefficient)
  __shared__ int   dstl[16];
  __shared__ float mu[16], rs[16];

  const int tid  = threadIdx.x;
  const int wid  = tid >> 5;
  const int lane = tid & 31;
  const int half = lane >> 4;
  const int mr   = lane & 15;
  const long long e0 = (long long)blockIdx.x * 16;

  __builtin_prefetch(pre_lin_w + tid, 0, 1);
  __builtin_prefetch(lin_edge_w + tid, 0, 1);

  for (int i = tid; i < 16 * 320; i += 256) {
    int e = i / 320, c = i - e * 320;
    long long ee = e0 + e; if (ee >= E) ee = E - 1;
    float v;
    if (c < 128)      v = node_in[(long long)edge_src[ee] * 128 + c];
    else if (c < 256) v = node_in[(long long)edge_dst[ee] * 128 + (c - 128)];
    else              v = xlen[ee * 64 + (c - 256)];
    Acat[i] = v;
  }
  if (tid < 16) {
    long long ee = e0 + tid;
    dstl[tid] = (ee < E) ? edge_dst[ee] : -1;
  }
  if (tid < 64) {
    int e = tid >> 2, h = tid & 3;
    long long ee = e0 + e;
    if (ee < E) {
      float a  = ws_alpha[ee * 4 + h];
      float sc = ws_scal[ee * 4 + h];
      long long ni = (long long)edge_dst[ee] * 4 + h;
      coefA[tid] = sc * a;
      coefN[tid] = sc * __expf(a - node_max[ni]) / (node_sum[ni] + 1e-16f);
    } else { coefA[tid] = 0.f; coefN[tid] = 0.f; }
  }
  __syncthreads();

  // msg = Acat @ pre_lin_w : [16x320]x[320x128], 8 waves x 16-col tiles
  {
    const int col0 = wid * 16;
    v8f acc = {0.f,0.f,0.f,0.f,0.f,0.f,0.f,0.f};
    acc = wmma_gemm_tile(Acat, 320, pre_lin_w, 128, col0, 320, acc);
#pragma unroll
    for (int r = 0; r < 8; ++r)
      msg[(r + 8 * half) * 128 + col0 + mr] = acc[r];
  }
  __syncthreads();

  // y0 = edge_out.reshape(16,512) @ lin_edge_w, A built on the fly:
  // edge_out[e][h*128+d] = msg[e][d] * coefA[e][h]
  if (wid < 4) {
    const int col0 = wid * 16;
    v8f acc = {0.f,0.f,0.f,0.f,0.f,0.f,0.f,0.f};
    for (int k0 = 0; k0 < 512; k0 += 4) {
      int ka = k0 + 2 * half;
      v2f a, b;
      a.x = msg[mr * 128 + (ka & 127)] * coefA[mr * 4 + (ka >> 7)];
      a.y = msg[mr * 128 + ((ka + 1) & 127)] * coefA[mr * 4 + ((ka + 1) >> 7)];
      b.x = lin_edge_w[ka * 64 + col0 + mr];
      b.y = lin_edge_w[(ka + 1) * 64 + col0 + mr];
      acc = WMMA_F32(acc, a, b);
    }
#pragma unroll
    for (int r = 0; r < 8; ++r)
      buf1[(r + 8 * half) * 64 + col0 + mr] = acc[r];
  }
  __syncthreads();

  // s-MLP: y1 = silu(LN(y0@s_w1+b1)); y2 = silu(LN(y1@s_w2+b2)); out = y2@s_w3+b3
  if (wid < 4) {
    const int col0 = wid * 16;
    v8f acc = {0.f,0.f,0.f,0.f,0.f,0.f,0.f,0.f};
    acc = wmma_gemm_tile(buf1, 64, s_w1, 64, col0, 64, acc);
#pragma unroll
    for (int r = 0; r < 8; ++r) {
      int m = r + 8 * half, n = col0 + mr;
      buf2[m * 64 + n] = acc[r] + s_b1[n];
    }
  }
  __syncthreads();
  ln_silu_16x64(buf2, s_g1, s_be1, mu, rs, tid, 256);
  if (wid < 4) {
    const int col0 = wid * 16;
    v8f acc = {0.f,0.f,0.f,0.f,0.f,0.f,0.f,0.f};
    acc = wmma_gemm_tile(buf2, 64, s_w2, 64, col0, 64, acc);
#pragma unroll
    for (int r = 0; r < 8; ++r) {
      int m = r + 8 * half, n = col0 + mr;
      buf1[m * 64 + n] = acc[r] + s_b2[n];
    }
  }
  __syncthreads();
  ln_silu_16x64(buf1, s_g2, s_be2, mu, rs, tid, 256);
  if (wid < 2) {  // [16x64]x[64x32]
    const int col0 = wid * 16;
    v8f acc = {0.f,0.f,0.f,0.f,0.f,0.f,0.f,0.f};
    acc = wmma_gemm_tile(buf1, 64, s_w3, 32, col0, 64, acc);
#pragma unroll
    for (int r = 0; r < 8; ++r) {
      int m = r + 8 * half;
      long long ee = e0 + m;
      if (ee < E) out_edge[ee * 32 + col0 + mr] = acc[r] + s_b3[col0 + mr];
    }
  }

  // node_fea[dst] += value * asm   (value = msg * scal; coefN = scal*asm)
  for (int i = tid; i < 16 * 512; i += 256) {
    int e = i >> 9, hd = i & 511, h = hd >> 7, d = hd & 127;
    long long ee = e0 + e;
    if (ee < E) {
      int dn = dstl[e];
      atomicAdd(&node_fea[(long long)dn * 512 + hd], msg[e * 128 + d] * coefN[e * 4 + h]);
    }
  }
}

// ---------------- Kernel 4: node_out = node_fea.reshape(N,512) @ lin_w ---------------------
__global__ __launch_bounds__(256) void node_out_kernel(
    const float* __restrict__ node_fea, const float* __restrict__ lin_w,
    float* __restrict__ out_node, long long Nn) {
  __shared__ float Af[16 * 512];
  const int tid  = threadIdx.x;
  const int wid  = tid >> 5;
  const int lane = tid & 31;
  const int half = lane >> 4;
  const int mr   = lane & 15;
  const long long n0 = (long long)blockIdx.x * 16;

  for (int i = tid; i < 16 * 512; i += 256) {
    int r = i >> 9, c = i & 511;
    long long nn = n0 + r; if (nn >= Nn) nn = Nn - 1;
    Af[i] = node_fea[nn * 512 + c];
  }
  __syncthreads();
  const int col0 = wid * 16;
  v8f acc = {0.f,0.f,0.f,0.f,0.f,0.f,0.f,0.f};
  acc = wmma_gemm_tile(Af, 512, lin_w, 128, col0, 512, acc);
#pragma unroll
  for (int r = 0; r < 8; ++r) {
    int m = r + 8 * half;
    long long nn = n0 + m;
    if (nn < Nn) out_node[nn * 128 + col0 + mr] = acc[r];
  }
}

extern "C" void kernel_launch(void* const* d_in, const int* in_sizes, int n_in,
                              void* d_out, int out_size, void* d_ws, size_t ws_size,
                              hipStream_t stream) {
  (void)n_in; (void)out_size; (void)ws_size;
  const float* node_in   = (const float*)d_in[0];
  // d_in[1] node_embed: unused by reference
  const float* edge_sh   = (const float*)d_in[2];
  const float* xlen      = (const float*)d_in[3];
  const int*   edge_src  = (const int*)d_in[4];
  const int*   edge_dst  = (const int*)d_in[5];
  // d_in[6] batch: unused
  const float* pre_lin_w = (const float*)d_in[7];
  const float* tp2_w1 = (const float*)d_in[8];
  const float* tp2_b1 = (const float*)d_in[9];
  const float* tp2_w2 = (const float*)d_in[10];
  const float* tp2_b2 = (const float*)d_in[11];
  const float* a_w1  = (const float*)d_in[12];
  const float* a_b1  = (const float*)d_in[13];
  const float* a_g1  = (const float*)d_in[14];
  const float* a_be1 = (const float*)d_in[15];
  const float* a_w2  = (const float*)d_in[16];
  const float* a_b2  = (const float*)d_in[17];
  const float* a_g2  = (const float*)d_in[18];
  const float* a_be2 = (const float*)d_in[19];
  const float* a_w3  = (const float*)d_in[20];
  const float* a_b3  = (const float*)d_in[21];
  const float* lin_w      = (const float*)d_in[22];
  const float* lin_edge_w = (const float*)d_in[23];
  const float* s_w1  = (const float*)d_in[24];
  const float* s_b1  = (const float*)d_in[25];
  const float* s_g1  = (const float*)d_in[26];
  const float* s_be1 = (const float*)d_in[27];
  const float* s_w2  = (const float*)d_in[28];
  const float* s_b2  = (const float*)d_in[29];
  const float* s_g2  = (const float*)d_in[30];
  const float* s_be2 = (const float*)d_in[31];
  const float* s_w3  = (const float*)d_in[32];
  const float* s_b3  = (const float*)d_in[33];

  const long long Nn = in_sizes[0] / 128;
  const long long E  = in_sizes[4];

  float* ws        = (float*)d_ws;
  float* ws_alpha  = ws;                       // [E,4]
  float* ws_scal   = ws_alpha + E * 4;         // [E,4]
  float* node_max  = ws_scal + E * 4;          // [N,4]
  float* node_sum  = node_max + Nn * 4;        // [N,4]
  float* node_fea  = node_sum + Nn * 4;        // [N,512]

  hipMemsetAsync(node_max, 0xFF, (size_t)(Nn * 4) * sizeof(float), stream);
  hipMemsetAsync(node_sum, 0,    (size_t)(Nn * 4) * sizeof(float), stream);
  hipMemsetAsync(node_fea, 0,    (size_t)(Nn * 512) * sizeof(float), stream);

  const unsigned etiles = (unsigned)((E + 15) / 16);
  edge_prep_kernel<<<etiles, 128, 0, stream>>>(
      xlen, edge_sh, edge_dst, tp2_w1, tp2_b1, tp2_w2, tp2_b2,
      a_w1, a_b1, a_g1, a_be1, a_w2, a_b2, a_g2, a_be2, a_w3, a_b3,
      ws_alpha, ws_scal, node_max, E);

  const long long EH = E * 4;
  softmax_sum_kernel<<<(unsigned)((EH + 255) / 256), 256, 0, stream>>>(
      ws_alpha, edge_dst, node_max, node_sum, EH);

  float* out_node = (float*)d_out;
  float* out_edge = out_node + Nn * 128;
  edge_main_kernel<<<etiles, 256, 0, stream>>>(
      node_in, xlen, edge_src, edge_dst, pre_lin_w, lin_edge_w,
      s_w1, s_b1, s_g1, s_be1, s_w2, s_b2, s_g2, s_be2, s_w3, s_b3,
      ws_alpha, ws_scal, node_max, node_sum, node_fea, out_edge, E);

  node_out_kernel<<<(unsigned)((Nn + 15) / 16), 256, 0, stream>>>(
      node_fea, lin_w, out_node, Nn);
}